// HTensorProductMLP_33809982554740
// MI455X (gfx1250) — compile-verified
//
#include <hip/hip_runtime.h>
#include <cmath>

// ---------------------------------------------------------------------------
// HTensorProductMLP for MI455X (gfx1250, wave32, WMMA)
//
// Roofline: main tensor product = ~2.8e10 fp32 MACs vs ~150MB HBM traffic
// (6.5us @ 23.3 TB/s) -> compute bound on the fp32 matrix pipe. Everything is
// cast as M=16(z) x K=32(v) x N=16(w) GEMMs on V_WMMA_F32_16X16X4_F32:
//   G_t[z,u,w] = sum_v W_t[u,v,w] * rhs_t[z,v]   (A = x2 data, B = W slice)
//   out       += lhs[z,u] * G_t[z,u,w]           (stage-B VALU FMA per u)
// B fragments (W slices) are z-independent: staged once per u into LDS in
// WMMA-fragment-swizzled layout and shared by both waves of the block.
// ---------------------------------------------------------------------------

typedef __attribute__((ext_vector_type(2))) float v2f;
typedef __attribute__((ext_vector_type(8))) float v8f;

#define Zn   32768
#define MULn 32
#define ROWL 128          // x1/x2 row length (4*MUL)
#define OUTL 224          // 32 + 96 + 96

__device__ __forceinline__ float gelu_phi(float x, float invPhi) {
    // jax.nn.gelu default (approximate=True, tanh form), then / PHI_NORM
    float x3 = x * x * x;
    float g = 0.5f * x * (1.0f + tanhf(0.79788456080286535588f * (x + 0.044715f * x3)));
    return g * invPhi;
}

// -------- kernel 1: h = phi(phi(emb@Wm0/8) @ Wm1 / sqrt(128)) --------------
__global__ void mlp_kernel(const float* __restrict__ emb,
                           const float* __restrict__ Wm0,
                           const float* __restrict__ Wm1,
                           float invPhi, float* __restrict__ h_out) {
    __shared__ float h1[128];
    int f = threadIdx.x;                    // 128 threads
    float t = 0.0f;
    for (int e = 0; e < 64; ++e) t += emb[e] * Wm0[e * 128 + f];
    h1[f] = gelu_phi(t * 0.125f, invPhi);   // 1/sqrt(64)
    __syncthreads();
    float t2 = 0.0f;
    for (int k = 0; k < 128; ++k) t2 += h1[k] * Wm1[k * 128 + f];
    h_out[f] = gelu_phi(t2 * 0.08838834764831845f, invPhi);  // 1/sqrt(128)
}

// -------- kernel 2: W_t[uvw] = coef_t * sum_x P_t[x,uvw] * h[x] ------------
// 84MB of P -> pure bandwidth; float4 coalesced. Coefficients folded:
//   t0=000:1/512  t1=011:1/512  t2=101:1/512  t3=110:1/(512*sqrt3)  t4=111:1/512
__global__ void reduce_w_kernel(const float* __restrict__ P000,
                                const float* __restrict__ P011,
                                const float* __restrict__ P101,
                                const float* __restrict__ P110,
                                const float* __restrict__ P111,
                                const float* __restrict__ h,
                                float* __restrict__ Wg) {
    __shared__ float hs[128];
    if (threadIdx.x < 128) hs[threadIdx.x] = h[threadIdx.x];
    __syncthreads();
    int gid = blockIdx.x * blockDim.x + threadIdx.x;   // 40960 threads, 4 elems each
    int t = gid / 8192;
    int e4 = (gid % 8192) * 4;
    const float* P = (t == 0) ? P000 : (t == 1) ? P011 : (t == 2) ? P101
                   : (t == 3) ? P110 : P111;
    const float coef = (t == 3) ? 0.0011276372445f : 0.001953125f;
    float4 acc = make_float4(0.f, 0.f, 0.f, 0.f);
    for (int x = 0; x < 128; ++x) {
        float4 p = *reinterpret_cast<const float4*>(P + (size_t)x * 32768 + e4);
        float hx = hs[x];
        acc.x += hx * p.x; acc.y += hx * p.y; acc.z += hx * p.z; acc.w += hx * p.w;
    }
    float4 r = make_float4(coef * acc.x, coef * acc.y, coef * acc.z, coef * acc.w);
    *reinterpret_cast<float4*>(Wg + (size_t)t * 32768 + e4) = r;
}

// -------- WMMA helper: D(16z x 16w) = sum over K=32 (v) of A x B -----------
// A 16x4 f32 layout: lanes 0-15 row M=lane   hold K = 4j+0, 4j+1
//                    lanes16-31 row M=lane-16 hold K = 4j+2, 4j+3
// B fragments are pre-swizzled in LDS so each lane reads one contiguous b64:
//   wsl[j*64 + lane*2 + {0,1}]
__device__ __forceinline__ v8f gemm_k32(const v2f* a, const float* wsl, int lane) {
    v8f c = {};
#pragma unroll
    for (int j = 0; j < 8; ++j) {
        v2f b = *reinterpret_cast<const v2f*>(wsl + j * 64 + lane * 2);
        c = __builtin_amdgcn_wmma_f32_16x16x4_f32(false, a[j], false, b,
                                                  (short)0, c, false, false);
    }
    return c;
}

// -------- kernel 3: main tensor product ------------------------------------
// Block = 2 waves (64 thr); wave w owns z-tile [zblk + 16w, +16).
// LDS: x1 tile 32x128, x2 tile 32x128, W slice 5*1024 (swizzled) = 52KB.
__global__ __launch_bounds__(64, 1)
void tp_kernel(const float* __restrict__ x1, const float* __restrict__ x2,
               const float* __restrict__ Wg, float* __restrict__ out) {
    extern __shared__ float smem[];
    float* x1t = smem;                 // 32*128
    float* x2t = x1t + 32 * ROWL;      // 32*128
    float* Wl  = x2t + 32 * ROWL;      // 5120 (per-u W slice, frag-swizzled)

    const int tid  = threadIdx.x;
    const int wave = tid >> 5;
    const int lane = tid & 31;
    const int zblk = blockIdx.x * 32;

    // stage x tiles (coalesced)
    for (int e = tid; e < 32 * ROWL; e += 64) {
        x1t[e] = x1[(size_t)zblk * ROWL + e];
        x2t[e] = x2[(size_t)zblk * ROWL + e];
    }
    __syncthreads();

    const int row = lane & 15;
    const int hi  = lane >> 4;

    // A fragments (resident whole kernel): s2 and v2[i], K=32 each
    v2f aS2[8], aV2[3][8];
    {
        const float* xr = x2t + (wave * 16 + row) * ROWL;
#pragma unroll
        for (int j = 0; j < 8; ++j) {
            int k0 = 4 * j + 2 * hi;
            v2f s; s.x = xr[k0]; s.y = xr[k0 + 1];
            aS2[j] = s;
#pragma unroll
            for (int i = 0; i < 3; ++i) {
                v2f v; v.x = xr[32 + 3 * k0 + i]; v.y = xr[32 + 3 * (k0 + 1) + i];
                aV2[i][j] = v;
            }
        }
    }

    // accumulators: 112 f32/lane (D layout: VGPR r -> z row r (lanes<16) / r+8)
    v8f o0[2] = {};
    v8f o1[3][2] = {};
    v8f o2[3][2] = {};

    for (int u = 0; u < 32; ++u) {
        __syncthreads();
        // stage W[t][u][v][w] -> B-fragment-swizzled LDS (coalesced global read)
        for (int e = tid; e < 5120; e += 64) {
            int t = e >> 10, r = e & 1023;
            int v = r >> 5, w = r & 31;
            int wt = w >> 4, nn = w & 15;
            int pair = v & 1, hh = (v >> 1) & 1, j = v >> 2;
            Wl[((t * 2 + wt) * 8 + j) * 64 + (hh * 16 + nn) * 2 + pair] =
                Wg[(size_t)t * 32768 + u * 1024 + r];
        }
        __syncthreads();

        // lhs values for this u (LDS broadcast per lane-half)
        float ls1[8], lv1[3][8];
#pragma unroll
        for (int r8 = 0; r8 < 8; ++r8) {
            const float* x1r = x1t + (wave * 16 + r8 + hi * 8) * ROWL;
            ls1[r8] = x1r[u];
#pragma unroll
            for (int i = 0; i < 3; ++i) lv1[i][r8] = x1r[32 + 3 * u + i];
        }

#pragma unroll
        for (int wt = 0; wt < 2; ++wt) {
            const float* W000 = Wl + (0 * 2 + wt) * 512;
            const float* W011 = Wl + (1 * 2 + wt) * 512;
            const float* W101 = Wl + (2 * 2 + wt) * 512;
            const float* W110 = Wl + (3 * 2 + wt) * 512;
            const float* W111 = Wl + (4 * 2 + wt) * 512;
            v8f g;

            // out0: s1 . (W000 @ s2)  +  sum_i v1_i . (W110 @ v2_i)
            g = gemm_k32(aS2, W000, lane);
#pragma unroll
            for (int r8 = 0; r8 < 8; ++r8) o0[wt][r8] += ls1[r8] * g[r8];
#pragma unroll
            for (int i = 0; i < 3; ++i) {
                g = gemm_k32(aV2[i], W110, lane);
#pragma unroll
                for (int r8 = 0; r8 < 8; ++r8) o0[wt][r8] += lv1[i][r8] * g[r8];
            }
            // out1_i: s1 . (W011 @ v2_i)  +  v1_i . (W101 @ s2)
#pragma unroll
            for (int i = 0; i < 3; ++i) {
                g = gemm_k32(aV2[i], W011, lane);
#pragma unroll
                for (int r8 = 0; r8 < 8; ++r8) o1[i][wt][r8] += ls1[r8] * g[r8];
            }
            g = gemm_k32(aS2, W101, lane);   // reused for all 3 components
#pragma unroll
            for (int i = 0; i < 3; ++i)
#pragma unroll
                for (int r8 = 0; r8 < 8; ++r8) o1[i][wt][r8] += lv1[i][r8] * g[r8];
            // out2: v1 x (W111 @ v2_j): eps_{i,j,k} with G_j
#pragma unroll
            for (int j3 = 0; j3 < 3; ++j3) {
                g = gemm_k32(aV2[j3], W111, lane);
                const int ka = (j3 + 1) % 3, kb = (j3 + 2) % 3;
#pragma unroll
                for (int r8 = 0; r8 < 8; ++r8) o2[ka][wt][r8] += lv1[kb][r8] * g[r8];
#pragma unroll
                for (int r8 = 0; r8 < 8; ++r8) o2[kb][wt][r8] -= lv1[ka][r8] * g[r8];
            }
        }
    }

    // writeback: row z = zblk + wave*16 + r8 + hi*8, col n = lane&15
#pragma unroll
    for (int r8 = 0; r8 < 8; ++r8) {
        float* orow = out + (size_t)(zblk + wave * 16 + r8 + hi * 8) * OUTL;
#pragma unroll
        for (int wt = 0; wt < 2; ++wt) {
            int w = wt * 16 + row;
            orow[w] = o0[wt][r8];
#pragma unroll
            for (int i = 0; i < 3; ++i) orow[32 + 3 * w + i] = o1[i][wt][r8];
#pragma unroll
            for (int k = 0; k < 3; ++k) orow[128 + 3 * w + k] = o2[k][wt][r8];
        }
    }
}

// ---------------------------------------------------------------------------
extern "C" void kernel_launch(void* const* d_in, const int* in_sizes, int n_in,
                              void* d_out, int out_size, void* d_ws, size_t ws_size,
                              hipStream_t stream) {
    const float* emb  = (const float*)d_in[0];
    const float* x1   = (const float*)d_in[1];
    const float* x2   = (const float*)d_in[2];
    const float* Wm0  = (const float*)d_in[3];
    const float* Wm1  = (const float*)d_in[4];
    const float* P000 = (const float*)d_in[5];
    const float* P011 = (const float*)d_in[6];
    const float* P101 = (const float*)d_in[7];
    const float* P110 = (const float*)d_in[8];
    const float* P111 = (const float*)d_in[9];
    float* outp = (float*)d_out;

    float* ws = (float*)d_ws;
    float* h  = ws;          // 128 floats
    float* Wg = ws + 128;    // 5 * 32768 floats

    // PHI_NORM via the same grid integral as the reference (tanh-gelu, N(0,1))
    double s = 0.0;
    for (int i = 0; i <= 200000; ++i) {
        double z = -10.0 + 1e-4 * (double)i;
        double pdf = exp(-z * z * 0.5) * 0.3989422804014327;
        double z3 = z * z * z;
        double g = 0.5 * z * (1.0 + tanh(0.79788456080286535588 * (z + 0.044715 * z3)));
        s += g * g * pdf;
    }
    float invPhi = (float)(1.0 / sqrt(s * 1e-4));

    mlp_kernel<<<1, 128, 0, stream>>>(emb, Wm0, Wm1, invPhi, h);
    reduce_w_kernel<<<160, 256, 0, stream>>>(P000, P011, P101, P110, P111, h, Wg);

    size_t shmem = (size_t)(2 * 32 * ROWL + 5120) * sizeof(float);  // 52 KB
    tp_kernel<<<Zn / 32, 64, shmem, stream>>>(x1, x2, Wg, outp);
}